// TNNCell_38749194944734
// MI455X (gfx1250) — compile-verified
//
#include <hip/hip_runtime.h>
#include <math.h>

// ---------------------------------------------------------------------------
// TNNCell for MI455X (gfx1250, wave32).
// Memory-bound (~76 MB traffic @ 23.3 TB/s ~ 3.3 us). The per-row MLPs are
// batched onto full-precision V_WMMA_F32_16X16X4_F32 tiles (16 rows / wave).
// All small operands live in zero-padded LDS so fragment construction and the
// activation epilogues are branchless (no per-iteration exec-mask churn).
// ---------------------------------------------------------------------------

typedef __attribute__((ext_vector_type(2))) float v2f;
typedef __attribute__((ext_vector_type(8))) float v8f;

#define SAMPLE_TIME 0.5f

__device__ __forceinline__ float fast_sigmoid(float x) {
    // 1/(1+exp(-x)) via v_exp_f32 + v_rcp_f32 (~1 ulp vs IEEE divide)
    return __builtin_amdgcn_rcpf(1.0f + __expf(-x));
}

__device__ __forceinline__ float fast_tanh(float x) {
#if __has_builtin(__builtin_amdgcn_tanhf)
    return __builtin_amdgcn_tanhf(x);   // gfx1250 v_tanh_f32
#else
    return tanhf(x);
#endif
}

__global__ __launch_bounds__(256) void tnn_cell_wmma_kernel(
    const float* __restrict__ inp,     // B x 7
    const float* __restrict__ hidden,  // B x 4
    const float* __restrict__ g_W,     // 11 x 15
    const float* __restrict__ g_b,     // 15
    const float* __restrict__ p_W1,    // 11 x 8
    const float* __restrict__ p_b1,    // 8
    const float* __restrict__ p_W2,    // 8 x 4
    const float* __restrict__ p_b2,    // 4
    const float* __restrict__ caps,    // 4
    const int*   __restrict__ adj,     // 4 x 6
    float* __restrict__ out_prev,      // B x 4
    float* __restrict__ out_new,       // B x 4
    int B)
{
    // --- activations: all_input = [inp(0..6) | hidden(7..10) | pad(11)=0] ---
    __shared__ float ldsAI[128][12];
    // --- per-wave WMMA staging (all 16-wide: unconditional stores) ---
    __shared__ float ldsC[8][16][16];  // |sigmoid| conducts (15 valid cols)
    __shared__ float ldsH[8][16][16];  // tanh hidden layer  (8 valid cols)
    __shared__ float ldsP[8][16][16];  // |power loss|       (4 valid cols)
    // --- zero-padded weights/biases (filled once per block) ---
    __shared__ float wGW[12][16];      // g_W  11x15 -> 12x16
    __shared__ float wP1[12][16];      // p_W1 11x8  -> 12x16
    __shared__ float wP2[8][16];       // p_W2 8x4   -> 8x16
    __shared__ float biasG[16], biasP1[16], biasP2[16];
    __shared__ int   ldsAdj[24];
    __shared__ float ldsICaps[4];

    const int tid  = threadIdx.x;
    const int wave = tid >> 5;
    const int ln   = tid & 31;
    const int nlo  = ln & 15;          // column / row-in-tile index
    const int hi   = ln >> 4;          // K half-select for fragments

    const int  base  = blockIdx.x * 128;
    const long inTot = (long)B * 7;

    // ---- stage weights (predicated, once per block, L2-resident) ----
    if (tid < 192) {                                   // 12x16 pads
        const int k = tid >> 4, n = tid & 15;
        wGW[k][n] = (k < 11 && n < 15) ? g_W[k * 15 + n] : 0.0f;
        wP1[k][n] = (k < 11 && n < 8)  ? p_W1[k * 8 + n] : 0.0f;
    }
    if (tid >= 128) {                                  // 8x16 pad (128 entries)
        const int i = tid - 128, k = i >> 4, n = i & 15;
        wP2[k][n] = (n < 4) ? p_W2[k * 4 + n] : 0.0f;
    }
    if (tid < 16) {
        biasG[tid]  = (tid < 15) ? g_b[tid]  : 0.0f;
        biasP1[tid] = (tid < 8)  ? p_b1[tid] : 0.0f;
        biasP2[tid] = (tid < 4)  ? p_b2[tid] : 0.0f;
    }
    if (tid >= 16 && tid < 40) ldsAdj[tid - 16]   = adj[tid - 16];
    if (tid >= 40 && tid < 44) ldsICaps[tid - 40] = __expf(caps[tid - 40]);

    // ---- stage activations: coalesced flat inp stream, b128 hidden ----
#pragma unroll
    for (int it = 0; it < 4; ++it) {                   // 896 = 3.5 * 256
        const int idx = tid + it * 256;
        if (idx < 896) {
            const long g = (long)base * 7 + idx;
            ldsAI[idx / 7][idx % 7] = (g < inTot) ? inp[g] : 0.0f;
        }
    }
    if (tid < 128) {
        const int  row  = base + tid;
        const long rowL = (row < B) ? (long)row : (long)(B - 1);
        const float4 h4 = *(const float4*)(hidden + rowL * 4);
        ldsAI[tid][7]  = h4.x;
        ldsAI[tid][8]  = h4.y;
        ldsAI[tid][9]  = h4.z;
        ldsAI[tid][10] = h4.w;
        ldsAI[tid][11] = 0.0f;                         // K=11 zero pad
    }
    __syncthreads();

    // ---- fragments (branchless ds_load; zero-padding lives in LDS) ----
    const int tileRow = wave * 16 + nlo;               // A-matrix row in block
    v2f a[3], bg[3], bp[3];
#pragma unroll
    for (int c = 0; c < 3; ++c) {
        const int k0 = 4 * c + 2 * hi;
        a[c].x  = ldsAI[tileRow][k0];
        a[c].y  = ldsAI[tileRow][k0 + 1];
        bg[c].x = wGW[k0][nlo];
        bg[c].y = wGW[k0 + 1][nlo];
        bp[c].x = wP1[k0][nlo];
        bp[c].y = wP1[k0 + 1][nlo];
    }

    // ---- GEMM stage 1: shared A, two fp32 accumulators ----
    v8f cg = {};
    v8f cp = {};
#pragma unroll
    for (int c = 0; c < 3; ++c) {
        cg = __builtin_amdgcn_wmma_f32_16x16x4_f32(false, a[c], false, bg[c],
                                                   (short)0, cg, false, false);
        cp = __builtin_amdgcn_wmma_f32_16x16x4_f32(false, a[c], false, bp[c],
                                                   (short)0, cp, false, false);
    }

    // Branchless epilogue in C/D layout: row = r + 8*hi, col = nlo.
    // Padded columns are exact zeros (B/bias pads) -> tanh(0)=0, |0|=0: safe.
    const float gbias  = biasG[nlo];
    const float p1bias = biasP1[nlo];
#pragma unroll
    for (int r = 0; r < 8; ++r) {
        const int m = r + 8 * hi;
        ldsC[wave][m][nlo] = fabsf(fast_sigmoid(cg[r] + gbias));
        ldsH[wave][m][nlo] = fast_tanh(cp[r] + p1bias);
    }
    __syncthreads();

    // ---- GEMM stage 2: tanh(16x8) @ p_W2(8x16 padded), K=8 -> 2 WMMA ----
    v2f a2[2], b2[2];
#pragma unroll
    for (int c = 0; c < 2; ++c) {
        const int k0 = 4 * c + 2 * hi;                 // always < 8
        a2[c].x = ldsH[wave][nlo][k0];
        a2[c].y = ldsH[wave][nlo][k0 + 1];
        b2[c].x = wP2[k0][nlo];
        b2[c].y = wP2[k0 + 1][nlo];
    }
    v8f cpl = {};
#pragma unroll
    for (int c = 0; c < 2; ++c) {
        cpl = __builtin_amdgcn_wmma_f32_16x16x4_f32(false, a2[c], false, b2[c],
                                                    (short)0, cpl, false, false);
    }
    const float p2bias = biasP2[nlo];
#pragma unroll
    for (int r = 0; r < 8; ++r) {
        ldsP[wave][r + 8 * hi][nlo] = fabsf(cpl[r] + p2bias);  // unconditional
    }
    __syncthreads();

    // ---- Finalize: adjacency diffusion + Euler step, 1 row / lane ----
    if (ln < 16) {
        const int row  = base + wave * 16 + ln;
        const int r128 = wave * 16 + ln;
        if (row < B) {
            const float h0 = ldsAI[r128][7],  h1 = ldsAI[r128][8];
            const float h2 = ldsAI[r128][9],  h3 = ldsAI[r128][10];
            const float h[4]     = {h0, h1, h2, h3};
            const float temps[6] = {h0, h1, h2, h3,
                                    ldsAI[r128][0], ldsAI[r128][1]};
            float o[4];
#pragma unroll
            for (int t = 0; t < 4; ++t) {
                float s = 0.0f;
#pragma unroll
                for (int j = 0; j < 6; ++j) {
                    const float g = ldsC[wave][ln][ldsAdj[t * 6 + j]];
                    s = fmaf(temps[j] - h[t], g, s);
                }
                const float v = h[t] + SAMPLE_TIME * ldsICaps[t] *
                                       (s + ldsP[wave][ln][t]);
                o[t] = fminf(3.0f, fmaxf(-1.0f, v));
            }
            *(float4*)(out_prev + (long)row * 4) = make_float4(h0, h1, h2, h3);
            *(float4*)(out_new  + (long)row * 4) = make_float4(o[0], o[1], o[2], o[3]);
        }
    }
}

extern "C" void kernel_launch(void* const* d_in, const int* in_sizes, int n_in,
                              void* d_out, int out_size, void* d_ws, size_t ws_size,
                              hipStream_t stream) {
    (void)n_in; (void)d_ws; (void)ws_size; (void)out_size;

    const float* inp    = (const float*)d_in[0];
    const float* hidden = (const float*)d_in[1];
    const float* g_W    = (const float*)d_in[2];
    const float* g_b    = (const float*)d_in[3];
    const float* p_W1   = (const float*)d_in[4];
    const float* p_b1   = (const float*)d_in[5];
    const float* p_W2   = (const float*)d_in[6];
    const float* p_b2   = (const float*)d_in[7];
    const float* caps   = (const float*)d_in[8];
    const int*   adj    = (const int*)d_in[9];

    const int B = in_sizes[0] / 7;                  // inp is B x 7
    float* out_prev = (float*)d_out;                // output 0: prev_out (B x 4)
    float* out_new  = (float*)d_out + (long)B * 4;  // output 1: clipped out (B x 4)

    const int rowsPerBlock = 128;                   // 8 waves x 16-row WMMA tiles
    const int blocks = (B + rowsPerBlock - 1) / rowsPerBlock;

    tnn_cell_wmma_kernel<<<blocks, 256, 0, stream>>>(
        inp, hidden, g_W, g_b, p_W1, p_b1, p_W2, p_b2, caps, adj,
        out_prev, out_new, B);
}